// attentiona_77000173682932
// MI455X (gfx1250) — compile-verified
//
#include <hip/hip_runtime.h>
#include <hip/hip_bf16.h>

#define DIMS  1024
#define NHEAD 16
#define HD    64
#define NB    4
#define NT    2048
#define E     16          // eff_span == HEAD == 16 tokens
#define NROWS (NB * E)    // 64 active rows

typedef __attribute__((ext_vector_type(2))) float v2f;
typedef __attribute__((ext_vector_type(8))) float v8f;

// ---------------------------------------------------------------------------
// V_WMMA_F32_16X16X4_F32 : D(16x16,f32) = A(16x4,f32) x B(4x16,f32) + C
// ---------------------------------------------------------------------------
__device__ __forceinline__ v8f wmma_f32(v2f a, v2f b, v8f c) {
  return __builtin_amdgcn_wmma_f32_16x16x4_f32(
      /*neg_a=*/false, a, /*neg_b=*/false, b,
      /*c_mod=*/(short)0, c, /*reuse_a=*/false, /*reuse_b=*/false);
}

// A fragment: A[m][k], rows m0..m0+15, cols kb..kb+3 of row-major src (ld)
// lane l holds k = kb + 2*(l>>4) + {0,1} for row m0 + (l&15)  -> contiguous f2
__device__ __forceinline__ v2f ldA(const float* src, int ld, int m0, int kb) {
  int l = threadIdx.x & 31;
  const float* p = src + (long)(m0 + (l & 15)) * ld + kb + ((l >> 4) << 1);
  return v2f{p[0], p[1]};
}

// B fragment for out = X @ W^T : B[k][n] = W[n][k], W row-major (ld)
// identical addressing pattern to ldA with n taking the role of m.
__device__ __forceinline__ v2f ldBT(const float* W, int ld, int n0, int kb) {
  int l = threadIdx.x & 31;
  const float* p = W + (long)(n0 + (l & 15)) * ld + kb + ((l >> 4) << 1);
  return v2f{p[0], p[1]};
}

// B fragment from row-major B[k][n] (ld): two rows strided by ld
__device__ __forceinline__ v2f ldB(const float* Bm, int ld, int kb, int n0) {
  int l = threadIdx.x & 31;
  const float* p = Bm + (long)(kb + ((l >> 4) << 1)) * ld + n0 + (l & 15);
  return v2f{p[0], p[ld]};
}

// ---------------------------------------------------------------------------
// Fill: out[b][t][:] = bo  (everything; active rows overwritten later)
// Also zeroes the diff accumulator used by the iterative loop.
// ---------------------------------------------------------------------------
__global__ void k_fill(float* __restrict__ out, const float* __restrict__ bo,
                       float* __restrict__ diffacc) {
  if (blockIdx.x == 0 && threadIdx.x == 0) *diffacc = 0.f;
  long idx = (long)blockIdx.x * blockDim.x + threadIdx.x;           // float4 idx
  const long tot = (long)NB * NT * DIMS / 4;
  if (idx < tot) {
    int n4 = (int)(idx & (DIMS / 4 - 1));
    ((float4*)out)[idx] = ((const float4*)bo)[n4];
  }
}

// ---------------------------------------------------------------------------
// LayerNorm of the 64 active x rows -> xn (64 x 1024)
// ---------------------------------------------------------------------------
__global__ void k_ln(const float* __restrict__ x, const float* __restrict__ w,
                     float* __restrict__ xn) {
  __shared__ float s1[256], s2[256];
  int r = blockIdx.x;                       // 0..63
  int b = r >> 4, t = r & 15;
  const float* xr = x + ((long)b * NT + t) * DIMS;
  float s = 0.f, q = 0.f;
  for (int c = threadIdx.x; c < DIMS; c += 256) { float vv = xr[c]; s += vv; q += vv * vv; }
  s1[threadIdx.x] = s; s2[threadIdx.x] = q;
  __syncthreads();
  for (int st = 128; st > 0; st >>= 1) {
    if ((int)threadIdx.x < st) { s1[threadIdx.x] += s1[threadIdx.x + st];
                                 s2[threadIdx.x] += s2[threadIdx.x + st]; }
    __syncthreads();
  }
  float m  = s1[0] / DIMS;
  float vr = s2[0] / DIMS - m * m;
  float rs = rsqrtf(vr + 1e-5f);
  for (int c = threadIdx.x; c < DIMS; c += 256)
    xn[(long)r * DIMS + c] = (xr[c] - m) * rs * w[c];
}

// ---------------------------------------------------------------------------
// QKV projection: (64 x 1024) @ W^T (+bias) * scale, scattered to (B,H,16,64)
// one wave per 16x16 output tile; grid (64 col-tiles, 4 row-tiles, 3 matrices)
// ---------------------------------------------------------------------------
__global__ void k_qkv(const float* __restrict__ xn,
                      const float* __restrict__ Wq, const float* __restrict__ bq,
                      const float* __restrict__ Wk,
                      const float* __restrict__ Wv, const float* __restrict__ bv,
                      float* __restrict__ q, float* __restrict__ k, float* __restrict__ v) {
  const int which = blockIdx.z;
  const float* W    = which == 0 ? Wq : (which == 1 ? Wk : Wv);
  const float* bias = which == 0 ? bq : (which == 1 ? nullptr : bv);
  const float  scl  = which == 2 ? 1.0f : 0.35355339059327373f;    // hd^-0.25
  float* dst = which == 0 ? q : (which == 1 ? k : v);

  int n0 = blockIdx.x * 16, m0 = blockIdx.y * 16;
  v8f acc = {};
  for (int kb = 0; kb < DIMS; kb += 4)
    acc = wmma_f32(ldA(xn, DIMS, m0, kb), ldBT(W, DIMS, n0, kb), acc);

  int l = threadIdx.x & 31, n = l & 15, hh = l >> 4;
  int col = n0 + n, h = col >> 6, d = col & 63;
  float badd = bias ? bias[col] : 0.f;
#pragma unroll
  for (int r = 0; r < 8; ++r) {
    int row = m0 + r + 8 * hh;                 // global active row 0..63
    int b = row >> 4, t = row & 15;
    dst[(((long)b * NHEAD + h) * E + t) * HD + d] = (acc[r] + badd) * scl;
  }
}

// ---------------------------------------------------------------------------
// Per-(b,h) helpers (one wave, LDS tiles of 16x64)
// ---------------------------------------------------------------------------
__device__ void proj64(const float* A /*LDS 16x64*/, const float* __restrict__ W,
                       const float* __restrict__ bias, float* D /*LDS 16x64*/) {
  __syncthreads();
  int l = threadIdx.x & 31, n = l & 15, hh = l >> 4;
  for (int n0 = 0; n0 < HD; n0 += 16) {
    v8f acc = {};
    for (int kb = 0; kb < HD; kb += 4)
      acc = wmma_f32(ldA(A, HD, 0, kb), ldBT(W, HD, n0, kb), acc);
    float bb = bias[n0 + n];
#pragma unroll
    for (int r = 0; r < 8; ++r) D[(r + 8 * hh) * HD + n0 + n] = acc[r] + bb;
  }
  __syncthreads();
}

__device__ void ln_rows(const float* src, const float* __restrict__ w,
                        float mult, float* dst) {
  __syncthreads();
  int l = threadIdx.x;
  if (l < 16) {
    float s = 0.f, q = 0.f;
    for (int d = 0; d < HD; ++d) { float xv = src[l * HD + d]; s += xv; q += xv * xv; }
    float m  = s / HD, vr = q / HD - m * m;
    float rs = rsqrtf(vr + 1e-5f) * mult;
    for (int d = 0; d < HD; ++d) dst[l * HD + d] = (src[l * HD + d] - m) * rs * w[d];
  }
  __syncthreads();
}

// ai(16x64) = softmax(qs @ ks^T * 1/8) @ vl
__device__ void attn16(const float* qs, const float* ks, const float* vl,
                       float* ai, float* ltile /*LDS 16x16*/) {
  int l = threadIdx.x, n = l & 15, hh = l >> 4;
  v8f acc = {};
  for (int kb = 0; kb < HD; kb += 4)
    acc = wmma_f32(ldA(qs, HD, 0, kb), ldBT(ks, HD, 0, kb), acc);
#pragma unroll
  for (int r = 0; r < 8; ++r) ltile[(r + 8 * hh) * 16 + n] = acc[r] * 0.125f;
  __syncthreads();
  if (l < 16) {
    float mx = -1e30f;
    for (int j = 0; j < 16; ++j) mx = fmaxf(mx, ltile[l * 16 + j]);
    float s = 0.f;
    for (int j = 0; j < 16; ++j) { float e = __expf(ltile[l * 16 + j] - mx);
                                   ltile[l * 16 + j] = e; s += e; }
    float inv = 1.f / s;
    for (int j = 0; j < 16; ++j) ltile[l * 16 + j] *= inv;
  }
  __syncthreads();
  for (int n0 = 0; n0 < HD; n0 += 16) {
    v8f a2 = {};
    for (int kb = 0; kb < 16; kb += 4)
      a2 = wmma_f32(ldA(ltile, 16, 0, kb), ldB(vl, HD, kb, n0), a2);
#pragma unroll
    for (int r = 0; r < 8; ++r) ai[(r + 8 * hh) * HD + n0 + n] = a2[r];
  }
  __syncthreads();
}

// ---------------------------------------------------------------------------
// Iterations 0 and 1 — one wave per (b,h). Emits qcur2 slice, kl1, vl1, ai1,
// and the global |ai1-ai0| sum (for the scalar early-exit decision).
// ---------------------------------------------------------------------------
__global__ void k_iter01(const float* __restrict__ q, const float* __restrict__ k,
                         const float* __restrict__ v,
                         const float* __restrict__ Wlq, const float* __restrict__ blq,
                         const float* __restrict__ Wlk, const float* __restrict__ blk,
                         const float* __restrict__ Wlv, const float* __restrict__ blv,
                         const float* __restrict__ lnc, const float* __restrict__ lnd,
                         const float* __restrict__ temp,
                         float* __restrict__ qcur2, float* __restrict__ kl_out,
                         float* __restrict__ vl_out, float* __restrict__ ai1_out,
                         float* __restrict__ diffacc) {
  __shared__ float qbuf[E * HD], kbuf[E * HD], vbuf[E * HD];
  __shared__ float ql[E * HD], kl[E * HD], vl[E * HD];
  __shared__ float qs[E * HD], ks[E * HD];
  __shared__ float ai[E * HD], prevb[E * HD];
  __shared__ float ltile[16 * 16];
  __shared__ float red;

  int bh = blockIdx.x;
  const float* qp = q + (long)bh * E * HD;
  const float* kp = k + (long)bh * E * HD;
  const float* vp = v + (long)bh * E * HD;
  for (int i = threadIdx.x; i < E * HD; i += 32) {
    qbuf[i] = qp[i]; kbuf[i] = kp[i]; vbuf[i] = vp[i];
  }
  __syncthreads();
  float t0 = *temp;

  // ---- iteration 0 ----
  proj64(qbuf, Wlq, blq, ql);
  proj64(kbuf, Wlk, blk, kl);
  proj64(vbuf, Wlv, blv, vl);
  float t_it = t0;
  float ts = (t_it != 1.0f && t_it > 0.f) ? rsqrtf(t_it) : 1.0f;
  ln_rows(ql, lnc, ts, qs);
  ln_rows(kl, lnd, 1.0f, ks);
  attn16(qs, ks, vl, ai, ltile);
  for (int i = threadIdx.x; i < E * HD; i += 32) {
    prevb[i] = ai[i];
    qbuf[i] += ai[i];          // qcur += iter_out (active slice)
    kbuf[i]  = kl[i];          // kcur = kl
    vbuf[i]  = vl[i];          // vcur = vl
  }
  __syncthreads();

  // ---- iteration 1 ----
  proj64(qbuf, Wlq, blq, ql);
  proj64(kbuf, Wlk, blk, kl);
  proj64(vbuf, Wlv, blv, vl);
  t_it = t0 + 0.005f;
  ts = (t_it != 1.0f && t_it > 0.f) ? rsqrtf(t_it) : 1.0f;
  ln_rows(ql, lnc, ts, qs);
  ln_rows(kl, lnd, 1.0f, ks);
  attn16(qs, ks, vl, ai, ltile);

  if (threadIdx.x == 0) red = 0.f;
  __syncthreads();
  float part = 0.f;
  for (int i = threadIdx.x; i < E * HD; i += 32) part += fabsf(ai[i] - prevb[i]);
  atomicAdd(&red, part);
  __syncthreads();
  if (threadIdx.x == 0) atomicAdd(diffacc, red);

  float* q2p = qcur2  + (long)bh * E * HD;
  float* klp = kl_out + (long)bh * E * HD;
  float* vlp = vl_out + (long)bh * E * HD;
  float* a1p = ai1_out + (long)bh * E * HD;
  for (int i = threadIdx.x; i < E * HD; i += 32) {
    q2p[i] = qbuf[i] + ai[i];
    klp[i] = kl[i];
    vlp[i] = vl[i];
    a1p[i] = ai[i];
  }
}

// ---------------------------------------------------------------------------
// Iteration 2 + scalar early-exit select; write merged (64 x 1024) attn matrix
// ---------------------------------------------------------------------------
__global__ void k_iter2(const float* __restrict__ qcur2, const float* __restrict__ kl1,
                        const float* __restrict__ vl1, const float* __restrict__ ai1,
                        const float* __restrict__ Wlq, const float* __restrict__ blq,
                        const float* __restrict__ Wlk, const float* __restrict__ blk,
                        const float* __restrict__ Wlv, const float* __restrict__ blv,
                        const float* __restrict__ lnc, const float* __restrict__ lnd,
                        const float* __restrict__ temp, const float* __restrict__ thr,
                        const float* __restrict__ fac, const float* __restrict__ diffacc,
                        float* __restrict__ attn_flat) {
  __shared__ float qbuf[E * HD], kbuf[E * HD], vbuf[E * HD];
  __shared__ float ql[E * HD], kl[E * HD], vl[E * HD];
  __shared__ float qs[E * HD], ks[E * HD];
  __shared__ float ai[E * HD], prevb[E * HD];
  __shared__ float ltile[16 * 16];

  int bh = blockIdx.x;
  for (int i = threadIdx.x; i < E * HD; i += 32) {
    qbuf[i] = qcur2[(long)bh * E * HD + i];
    kbuf[i] = kl1  [(long)bh * E * HD + i];
    vbuf[i] = vl1  [(long)bh * E * HD + i];
    prevb[i] = ai1 [(long)bh * E * HD + i];
  }
  __syncthreads();

  float diff1 = *diffacc / (float)((long)NB * NHEAD * NT * HD);
  bool done1 = diff1 < (*thr + *fac * diff1);

  proj64(qbuf, Wlq, blq, ql);
  proj64(kbuf, Wlk, blk, kl);
  proj64(vbuf, Wlv, blv, vl);
  float t_it = *temp + 0.01f;
  float ts = (t_it != 1.0f && t_it > 0.f) ? rsqrtf(t_it) : 1.0f;
  ln_rows(ql, lnc, ts, qs);
  ln_rows(kl, lnd, 1.0f, ks);
  attn16(qs, ks, vl, ai, ltile);

  int b = bh >> 4, h = bh & 15;
  for (int i = threadIdx.x; i < E * HD; i += 32) {
    int t = i >> 6, d = i & 63;
    float val = done1 ? prevb[i] : ai[i];
    attn_flat[((long)(b * E + t)) * DIMS + h * HD + d] = val;
  }
}

// ---------------------------------------------------------------------------
// Final projection: out[active rows] = attn_flat @ Wo^T + bo
// ---------------------------------------------------------------------------
__global__ void k_outproj(const float* __restrict__ attn_flat, const float* __restrict__ Wo,
                          const float* __restrict__ bo, float* __restrict__ out) {
  int n0 = blockIdx.x * 16, m0 = blockIdx.y * 16;
  v8f acc = {};
  for (int kb = 0; kb < DIMS; kb += 4)
    acc = wmma_f32(ldA(attn_flat, DIMS, m0, kb), ldBT(Wo, DIMS, n0, kb), acc);
  int l = threadIdx.x, n = l & 15, hh = l >> 4;
  float bb = bo[n0 + n];
#pragma unroll
  for (int r = 0; r < 8; ++r) {
    int row = m0 + r + 8 * hh;
    int b = row >> 4, t = row & 15;
    out[((long)b * NT + t) * DIMS + n0 + n] = acc[r] + bb;
  }
}

// ---------------------------------------------------------------------------
extern "C" void kernel_launch(void* const* d_in, const int* in_sizes, int n_in,
                              void* d_out, int out_size, void* d_ws, size_t ws_size,
                              hipStream_t stream) {
  (void)in_sizes; (void)n_in; (void)out_size; (void)ws_size;
  const float* x   = (const float*)d_in[0];
  const float* Wq  = (const float*)d_in[1];
  const float* bq  = (const float*)d_in[2];
  const float* Wk  = (const float*)d_in[3];
  const float* Wv  = (const float*)d_in[4];
  const float* bv  = (const float*)d_in[5];
  const float* Wo  = (const float*)d_in[6];
  const float* bo  = (const float*)d_in[7];
  const float* lna = (const float*)d_in[8];
  const float* lnc = (const float*)d_in[9];
  const float* lnd = (const float*)d_in[10];
  const float* Wlq = (const float*)d_in[11];
  const float* blq = (const float*)d_in[12];
  const float* Wlk = (const float*)d_in[13];
  const float* blk = (const float*)d_in[14];
  const float* Wlv = (const float*)d_in[15];
  const float* blv = (const float*)d_in[16];
  const float* temp= (const float*)d_in[17];
  const float* thr = (const float*)d_in[18];
  const float* fac = (const float*)d_in[19];
  float* out = (float*)d_out;
  float* ws  = (float*)d_ws;

  const long SEG = (long)NROWS * DIMS;  // 65536 floats
  float* xn      = ws;
  float* q       = ws + 1 * SEG;
  float* k       = ws + 2 * SEG;
  float* v       = ws + 3 * SEG;
  float* qcur2   = ws + 4 * SEG;
  float* kl1     = ws + 5 * SEG;
  float* vl1     = ws + 6 * SEG;
  float* ai1     = ws + 7 * SEG;
  float* aflat   = ws + 8 * SEG;
  float* diffacc = ws + 9 * SEG;

  // 1) broadcast bo over the entire output (t>=16 rows are exactly bo)
  k_fill<<<dim3((NB * NT * DIMS / 4) / 256), dim3(256), 0, stream>>>(out, bo, diffacc);
  // 2) LN of the 64 active rows
  k_ln<<<dim3(NROWS), dim3(256), 0, stream>>>(x, lna, xn);
  // 3) QKV projections (f32 WMMA)
  k_qkv<<<dim3(DIMS / 16, NROWS / 16, 3), dim3(32), 0, stream>>>(
      xn, Wq, bq, Wk, Wv, bv, q, k, v);
  // 4) iterations 0+1, global diff accumulation
  k_iter01<<<dim3(NB * NHEAD), dim3(32), 0, stream>>>(
      q, k, v, Wlq, blq, Wlk, blk, Wlv, blv, lnc, lnd, temp,
      qcur2, kl1, vl1, ai1, diffacc);
  // 5) iteration 2 + early-exit select
  k_iter2<<<dim3(NB * NHEAD), dim3(32), 0, stream>>>(
      qcur2, kl1, vl1, ai1, Wlq, blq, Wlk, blk, Wlv, blv, lnc, lnd,
      temp, thr, fac, diffacc, aflat);
  // 6) output projection for active rows
  k_outproj<<<dim3(DIMS / 16, NROWS / 16), dim3(32), 0, stream>>>(aflat, Wo, bo, out);
}